// cnn_no_pad_35364760715453
// MI455X (gfx1250) — compile-verified
//
#include <hip/hip_runtime.h>
#include <hip/hip_bf16.h>

#define B_    32
#define L_    16384
#define D_    10
#define KS_   22
#define TOPK_ 50
#define H_    2
#define NIMG  (B_ * 2)
#define KDIM  (KS_ * D_)          // 220 = effective GEMM K
#define TILE_M 256                // per block; each of 8 waves does 32 rows
#define HALO   (KS_ - 1)          // 21
#define LDSROWS (TILE_M + HALO)   // 277

typedef __attribute__((ext_vector_type(2))) float v2f;
typedef __attribute__((ext_vector_type(8))) float v8f;

// ---------------------------------------------------------------------------
// 1) Embedding gather: X[img, l, d] = emb[tok[img, l], d]
// ---------------------------------------------------------------------------
__global__ void embed_kernel(const int* __restrict__ tok,
                             const float* __restrict__ emb,
                             float* __restrict__ X) {
  int idx = blockIdx.x * blockDim.x + threadIdx.x;   // over NIMG*L_
  if (idx >= NIMG * L_) return;
  int t = tok[idx];                                  // [B,2,L] flat == [img, l]
  const float* er = emb + t * D_;
  float* xr = X + idx * D_;
#pragma unroll
  for (int d = 0; d < D_; ++d) xr[d] = er[d];
}

// ---------------------------------------------------------------------------
// 2) Conv as strided-window GEMM via V_WMMA_F32_16X16X4_F32.
//    out[i, n] = relu(bias + sum_k flat_in[(i-10)*10 + k] * Wfull[n, k])
//    Wfull[n, dy*10+x] = w[dy, x - n + 10]   (always in-bounds for n<10)
//    Each wave: 32x16 output tile = 2 accumulators sharing one B fragment.
// ---------------------------------------------------------------------------
__global__ void __launch_bounds__(256)
conv_wmma_kernel(const float* __restrict__ Xin, float* __restrict__ Xout,
                 const float* __restrict__ w, const float* __restrict__ bias) {
  __shared__ float s_in[LDSROWS * D_];     // 2770 floats (tile + halo, zero pad)
  __shared__ float s_w[(KDIM / 2) * 16 * 2]; // 3520: [pair p][n][j] = Wfull[n, 2p+j]

  const int tile = blockIdx.x;             // 0..63
  const int img  = blockIdx.y;             // 0..63
  const int tid  = threadIdx.x;
  const float* inimg = Xin + img * (L_ * D_);

  // CDNA5 prefetch of next tile's input window (global_prefetch_b8)
  if (tile + 1 < (int)gridDim.x && tid == 0) {
    __builtin_prefetch(inimg + ((tile + 1) * TILE_M - 10) * D_, 0, 3);
  }

  // Pair-interleaved weights: s_w[(p*16 + n)*2 + j] = Wfull[n, 2p+j]
  for (int i = tid; i < (KDIM / 2) * 16 * 2; i += 256) {
    int j = i & 1;
    int pn = i >> 1;
    int n = pn & 15;
    int k = ((pn >> 4) << 1) + j;
    float v = 0.0f;
    if (n < D_) {
      int dy = k / D_, x = k % D_;
      v = w[dy * KS_ + (x - n + 10)];      // kernel col in [1,19], always valid
    }
    s_w[i] = v;
  }
  // Stage input rows [tile*256-10, tile*256+267): flat copy, zero padded
  const int f0 = (tile * TILE_M - 10) * D_;
  for (int i = tid; i < LDSROWS * D_; i += 256) {
    int f = f0 + i;
    s_in[i] = (f >= 0 && f < L_ * D_) ? inimg[f] : 0.0f;
  }
  __syncthreads();

  const int lane = tid & 31;
  const int wave = tid >> 5;
  const int half = lane >> 4;              // 0: lanes 0-15, 1: lanes 16-31
  const int mn   = lane & 15;              // M for A-frag, N for B/D-frag
  // A[m, k] = s_in[m*10 + k]; two M-tiles per wave
  const float* arow0 = s_in + (wave * 32 + mn) * D_;
  const float* arow1 = arow0 + 16 * D_;

  v8f acc0 = {};
  v8f acc1 = {};
  for (int k0 = 0; k0 < KDIM; k0 += 4) {
    int ka = k0 + 2 * half;                // A/B VGPR0 holds K=ka, VGPR1 K=ka+1
    v2f a0 = *(const v2f*)(arow0 + ka);    // 8B aligned: even index
    v2f a1 = *(const v2f*)(arow1 + ka);
    v2f b  = *(const v2f*)(s_w + (((k0 >> 1) + half) * 16 + mn) * 2);
    acc0 = __builtin_amdgcn_wmma_f32_16x16x4_f32(
        false, a0, false, b, (short)0, acc0, false, false);
    acc1 = __builtin_amdgcn_wmma_f32_16x16x4_f32(
        false, a1, false, b, (short)0, acc1, false, false);
  }

  // fused bias + relu + store (only N=0..9 are real columns)
  const float bv = bias[0];
  if (mn < D_) {
    float* outimg = Xout + img * (L_ * D_);
    int rbase = tile * TILE_M + wave * 32 + half * 8;  // D VGPR r -> row r+8*half
#pragma unroll
    for (int r = 0; r < 8; ++r) {
      float v0 = acc0[r] + bv;
      float v1 = acc1[r] + bv;
      v0 = v0 > 0.0f ? v0 : 0.0f;
      v1 = v1 > 0.0f ? v1 : 0.0f;
      outimg[(rbase + r) * D_ + mn] = v0;
      outimg[(rbase + 16 + r) * D_ + mn] = v1;
    }
  }
}

// ---------------------------------------------------------------------------
// 3) Top-50 per (img, channel): whole 16384-row cached in 64KB dynamic LDS
//    (CDNA5 WGP has 320KB), 50 packed-key max extractions.
// ---------------------------------------------------------------------------
__global__ void __launch_bounds__(256)
topk_kernel(const float* __restrict__ Xin, float* __restrict__ F) {
  extern __shared__ unsigned char s_dyn[];
  float* s_vals = (float*)s_dyn;                                   // 16384
  unsigned long long* s_red =
      (unsigned long long*)(s_dyn + L_ * sizeof(float));           // 256

  const int img = blockIdx.x;   // 0..63
  const int d   = blockIdx.y;   // 0..9
  const int tid = threadIdx.x;  // 256
  const float* src = Xin + img * (L_ * D_) + d;
  for (int i = tid; i < L_; i += 256) s_vals[i] = src[i * D_];
  __syncthreads();

  for (int t = 0; t < TOPK_; ++t) {
    unsigned long long best = 0ull;
    for (int i = tid; i < L_; i += 256) {
      unsigned int u = __float_as_uint(s_vals[i]);
      unsigned int key = (u & 0x80000000u) ? ~u : (u | 0x80000000u);  // order-preserving
      unsigned long long p = ((unsigned long long)key << 32) | (unsigned int)i;
      best = p > best ? p : best;
    }
    s_red[tid] = best;
    __syncthreads();
    for (int s = 128; s > 0; s >>= 1) {
      if (tid < s) {
        if (s_red[tid + s] > s_red[tid]) s_red[tid] = s_red[tid + s];
      }
      __syncthreads();
    }
    if (tid == 0) {
      unsigned long long bp = s_red[0];
      unsigned int pos = (unsigned int)bp;
      unsigned int key = (unsigned int)(bp >> 32);
      unsigned int u = (key & 0x80000000u) ? (key ^ 0x80000000u) : ~key;
      F[img * (TOPK_ * D_) + t * D_ + d] = __uint_as_float(u);  // sorted desc
      s_vals[pos] = -__builtin_huge_valf();
    }
    __syncthreads();
  }
}

// ---------------------------------------------------------------------------
// 4) Two-head cross attention (both directions) + sum over queries + head.
//    Tiny (50x50), one block per batch element.
// ---------------------------------------------------------------------------
__global__ void __launch_bounds__(128)
attn_kernel(const float* __restrict__ F,
            const float* __restrict__ wq, const float* __restrict__ bq,
            const float* __restrict__ wk, const float* __restrict__ bk,
            const float* __restrict__ wv, const float* __restrict__ bv,
            const float* __restrict__ wo, const float* __restrict__ bo,
            const float* __restrict__ wf, const float* __restrict__ bf,
            float* __restrict__ out) {
  __shared__ float sf1[TOPK_ * D_], sf2[TOPK_ * D_];
  __shared__ float sQ[TOPK_ * H_ * D_], sK[TOPK_ * H_ * D_], sV[TOPK_ * H_ * D_];
  __shared__ float sA[H_ * TOPK_ * TOPK_];
  __shared__ float sO[TOPK_ * H_ * D_];
  __shared__ float sr[2][D_];

  const int b = blockIdx.x, tid = threadIdx.x;
  for (int i = tid; i < TOPK_ * D_; i += 128) {
    sf1[i] = F[(b * 2 + 0) * (TOPK_ * D_) + i];
    sf2[i] = F[(b * 2 + 1) * (TOPK_ * D_) + i];
  }
  __syncthreads();

  for (int dir = 0; dir < 2; ++dir) {
    const float* qf = dir ? sf2 : sf1;
    const float* kv = dir ? sf1 : sf2;
    // Q,K,V projections: [s][h][e]
    for (int i = tid; i < TOPK_ * H_ * D_; i += 128) {
      int s = i / (H_ * D_);
      int he = i % (H_ * D_);
      float aq = bq[he], ak = bk[he], av = bv[he];
      for (int dd = 0; dd < D_; ++dd) {
        float xq = qf[s * D_ + dd], xk = kv[s * D_ + dd];
        aq += xq * wq[dd * H_ * D_ + he];
        ak += xk * wk[dd * H_ * D_ + he];
        av += xk * wv[dd * H_ * D_ + he];
      }
      sQ[i] = aq; sK[i] = ak; sV[i] = av;
    }
    __syncthreads();
    // logits + softmax, one (h,q) row per thread
    const float scale = 0.31622776601683794f;   // 1/sqrt(10)
    for (int row = tid; row < H_ * TOPK_; row += 128) {
      int h = row / TOPK_, qq = row % TOPK_;
      float tmp[TOPK_];
      float mx = -3.4e38f;
      for (int ss = 0; ss < TOPK_; ++ss) {
        float acc = 0.0f;
        for (int e = 0; e < D_; ++e)
          acc += sQ[(qq * H_ + h) * D_ + e] * sK[(ss * H_ + h) * D_ + e];
        acc *= scale;
        tmp[ss] = acc;
        mx = acc > mx ? acc : mx;
      }
      float sum = 0.0f;
      for (int ss = 0; ss < TOPK_; ++ss) { float e = __expf(tmp[ss] - mx); tmp[ss] = e; sum += e; }
      float inv = 1.0f / sum;
      for (int ss = 0; ss < TOPK_; ++ss) sA[(h * TOPK_ + qq) * TOPK_ + ss] = tmp[ss] * inv;
    }
    __syncthreads();
    // O = A @ V
    for (int i = tid; i < TOPK_ * H_ * D_; i += 128) {
      int qq = i / (H_ * D_);
      int he = i % (H_ * D_);
      int h = he / D_, e = he % D_;
      float acc = 0.0f;
      for (int ss = 0; ss < TOPK_; ++ss)
        acc += sA[(h * TOPK_ + qq) * TOPK_ + ss] * sV[(ss * H_ + h) * D_ + e];
      sO[i] = acc;
    }
    __syncthreads();
    // output projection + sum over queries
    for (int dd = tid; dd < D_; dd += 128) {
      float acc = (float)TOPK_ * bo[dd];
      for (int qq = 0; qq < TOPK_; ++qq)
        for (int h = 0; h < H_; ++h)
          for (int e = 0; e < D_; ++e)
            acc += sO[(qq * H_ + h) * D_ + e] * wo[(h * D_ + e) * D_ + dd];
      sr[dir][dd] = acc;
    }
    __syncthreads();
  }

  // final dense [10 -> 2] + softmax
  if (tid < 2) {
    int dir = tid;
    float z0 = bf[0], z1 = bf[1];
    for (int dd = 0; dd < D_; ++dd) {
      z0 += sr[dir][dd] * wf[dd * 2 + 0];
      z1 += sr[dir][dd] * wf[dd * 2 + 1];
    }
    float m = z0 > z1 ? z0 : z1;
    float e0 = __expf(z0 - m), e1 = __expf(z1 - m);
    float s = e0 + e1;
    out[(b * 2 + dir) * 2 + 0] = e0 / s;
    out[(b * 2 + dir) * 2 + 1] = e1 / s;
  }
}

// ---------------------------------------------------------------------------
extern "C" void kernel_launch(void* const* d_in, const int* in_sizes, int n_in,
                              void* d_out, int out_size, void* d_ws, size_t ws_size,
                              hipStream_t stream) {
  const int*   tok = (const int*)  d_in[0];
  const float* emb = (const float*)d_in[1];
  const float* w1  = (const float*)d_in[2];
  const float* b1  = (const float*)d_in[3];
  const float* w2  = (const float*)d_in[4];
  const float* b2  = (const float*)d_in[5];
  const float* w3  = (const float*)d_in[6];
  const float* b3  = (const float*)d_in[7];
  const float* wq  = (const float*)d_in[8];
  const float* bq  = (const float*)d_in[9];
  const float* wk  = (const float*)d_in[10];
  const float* bk  = (const float*)d_in[11];
  const float* wv  = (const float*)d_in[12];
  const float* bv  = (const float*)d_in[13];
  const float* wo  = (const float*)d_in[14];
  const float* bo  = (const float*)d_in[15];
  const float* wf  = (const float*)d_in[16];
  const float* bf  = (const float*)d_in[17];
  float* out = (float*)d_out;

  const size_t imgElems = (size_t)NIMG * L_ * D_;     // 10,485,760 floats
  float* X0 = (float*)d_ws;
  float* X1 = X0 + imgElems;
  float* F  = X1 + imgElems;                          // 64*50*10 floats

  embed_kernel<<<dim3((NIMG * L_ + 255) / 256), 256, 0, stream>>>(tok, emb, X0);

  dim3 cgrid(L_ / TILE_M, NIMG);
  conv_wmma_kernel<<<cgrid, 256, 0, stream>>>(X0, X1, w1, b1);
  conv_wmma_kernel<<<cgrid, 256, 0, stream>>>(X1, X0, w2, b2);
  conv_wmma_kernel<<<cgrid, 256, 0, stream>>>(X0, X1, w3, b3);

  size_t topk_lds = (size_t)L_ * sizeof(float) + 256 * sizeof(unsigned long long);
  topk_kernel<<<dim3(NIMG, D_), 256, topk_lds, stream>>>(X1, F);

  attn_kernel<<<B_, 128, 0, stream>>>(F, wq, bq, wk, bk, wv, bv, wo, bo, wf, bf, out);
}